// SymmetricTensorProduct_70583492542672
// MI455X (gfx1250) — compile-verified
//
#include <hip/hip_runtime.h>

// ---------------------------------------------------------------------------
// SymmetricTensorProduct for CDNA5 / gfx1250 (wave32, WMMA).
//
// out[n,c,i] = sum_f F[n,c,f] * W[c][f,i]
// Features (900 of them, branch-free): with xe = [x0..x8, 1],
//   f = a*100 + rr,  a in [0,9),  rr = b*10 + j,  b,j in [0,10)
//   feat(f) = xe[a] * PP[rr],  PP[rr] = xe[b]*xe[j]
// covering cubics (b,j<9), quads (b<9, j==9), linears (b==9==j); redundant
// slots get zero weight in W. W folds u/w/(1/mul) per channel (kernel 1).
// Kernel 2: per-channel GEMM with v_wmma_f32_16x16x4_f32, K=900 (225 chunks).
// ---------------------------------------------------------------------------

typedef float v2f __attribute__((ext_vector_type(2)));
typedef float v8f __attribute__((ext_vector_type(8)));

#define NB 1024
#define CB 128
#define DB 9
#define NFP 900         // padded feature count (9 a-blocks x 100)
#define NCOL 10         // stored W columns: 9 outputs + 1 zero column
#define WCH 9000        // floats per channel in W buffer: (NFP/2)*NCOL*2
#define BLK 128         // threads per block (4 waves)
#define NROW 64         // n rows per block
#define NCH 25          // K-chunks per a-block (100 features / 4)

// sPP layout: pair p (=rr>>1, 0..49), odd p offset +32 dwords for bank split:
//   dword index = p*128 + (p&1)*32 + nloc*2 + (rr&1)
#define SPP_DWORDS (50 * 128 + 32)

// ---------------------------------------------------------------------------
// Kernel 1: W[c*WCH + (f>>1)*20 + i*2 + (f&1)] = folded weight (0 on padding)
// ---------------------------------------------------------------------------
__global__ __launch_bounds__(256)
void build_weights(const float* __restrict__ u1_0e, const float* __restrict__ w1_0e,
                   const float* __restrict__ u1_1o, const float* __restrict__ w1_1o,
                   const float* __restrict__ u1_2e, const float* __restrict__ w1_2e,
                   const float* __restrict__ u2_0e, const float* __restrict__ w2_0e,
                   const float* __restrict__ u2_1o, const float* __restrict__ w2_1o,
                   const float* __restrict__ u2_2e, const float* __restrict__ w2_2e,
                   const float* __restrict__ u3_0e, const float* __restrict__ w3_0e,
                   const float* __restrict__ u3_1o, const float* __restrict__ w3_1o,
                   const float* __restrict__ u3_2e, const float* __restrict__ w3_2e,
                   float* __restrict__ W)
{
    int tid = blockIdx.x * blockDim.x + threadIdx.x;
    if (tid >= CB * NFP * NCOL) return;
    int c = tid / (NFP * NCOL);
    int r = tid % (NFP * NCOL);
    int f = r / NCOL;
    int i = r % NCOL;

    int a  = f / 100;
    int rr = f % 100;
    int b  = rr / 10;
    int j  = rr % 10;

    float val = 0.0f;
    if (i < 9) {
        // irrep selection from output column i (concat order: 0e | 1o | 2e)
        int irsel, iloc, dim;
        if (i == 0)      { irsel = 0; iloc = 0;     dim = 1; }
        else if (i < 4)  { irsel = 1; iloc = i - 1; dim = 3; }
        else             { irsel = 2; iloc = i - 4; dim = 5; }

        const float* u = nullptr; const float* w = nullptr;
        int mul = 0, base = 0; bool ok = true;
        if (b < 9 && j < 9) {            // cubic: C3[a,b,j]
            u = (irsel == 0) ? u3_0e : (irsel == 1) ? u3_1o : u3_2e;
            w = (irsel == 0) ? w3_0e : (irsel == 1) ? w3_1o : w3_2e;
            mul = (irsel == 0) ? 4 : (irsel == 1) ? 7 : 10;
            base = (a * 9 + b) * 9 + j;
        } else if (b < 9 && j == 9) {    // quadratic: C2[a,b]
            u = (irsel == 0) ? u2_0e : (irsel == 1) ? u2_1o : u2_2e;
            w = (irsel == 0) ? w2_0e : (irsel == 1) ? w2_1o : w2_2e;
            mul = (irsel == 2) ? 3 : 2;
            base = a * 9 + b;
        } else if (b == 9 && j == 9) {   // linear: C1[a]
            u = (irsel == 0) ? u1_0e : (irsel == 1) ? u1_1o : u1_2e;
            w = (irsel == 0) ? w1_0e : (irsel == 1) ? w1_1o : w1_2e;
            mul = 1;
            base = a;
        } else {
            ok = false;                  // (b==9, j<9): redundant slot
        }
        if (ok) {
            float acc = 0.0f;
            for (int k = 0; k < mul; ++k)
                acc += u[(base * mul + k) * dim + iloc] * w[k * CB + c];
            val = acc / (float)mul;
        }
    }
    W[c * WCH + (f >> 1) * (NCOL * 2) + i * 2 + (f & 1)] = val;
}

// ---------------------------------------------------------------------------
// Kernel 2: per-channel GEMM  out[n, 0:9] = F[n, 0:900] @ W[c]
// grid = (NB/NROW, CB); block = 128 (4 waves); each wave owns a 16-n tile.
// ---------------------------------------------------------------------------
__global__ __launch_bounds__(BLK)
void stp_main(const float* __restrict__ x, const float* __restrict__ W,
              float* __restrict__ out)
{
    __shared__ float sW[NFP * NCOL];   // 36,000 B: [g=f>>1][10 cols][2]
    __shared__ float sPP[SPP_DWORDS];  // 25,728 B: pair products per row-n

    const int c    = blockIdx.y;
    const int tidx = threadIdx.x;

    // Stage W[c] (coalesced, layout-preserving copy).
    {
        const float* Wc = W + (size_t)c * WCH;
        for (int t = tidx; t < NFP * NCOL; t += BLK) sW[t] = Wc[t];
    }

    const int lane = tidx & 31;
    const int wave = tidx >> 5;
    const int m    = lane & 15;          // A: row within tile; B/D: column
    const int hi   = lane >> 4;          // K half-select within fragments
    const int nloc = wave * 16 + m;      // row index within block [0, 64)
    const int n0   = blockIdx.x * NROW + wave * 16;
    const int n    = n0 + m;

    // xe = [x(n,c,0..8), 1]
    float xe[10];
    {
        const float* xp = x + ((size_t)n * CB + c) * DB;
        #pragma unroll
        for (int d = 0; d < DB; ++d) xe[d] = xp[d];
        xe[9] = 1.0f;
    }

    // Build PP[rr] = xe[b]*xe[j] (both lane halves write identical values).
    #pragma unroll
    for (int b = 0; b < 10; ++b) {
        #pragma unroll
        for (int jp = 0; jp < 5; ++jp) {
            const int p = b * 5 + jp;                 // rr>>1, rr = b*10+2*jp
            v2f v; v.x = xe[b] * xe[2 * jp]; v.y = xe[b] * xe[2 * jp + 1];
            *(v2f*)&sPP[p * 128 + (p & 1) * 32 + nloc * 2] = v;
        }
    }
    __syncthreads();

    const int mc = (m < 9) ? m : 9;      // lanes 9..15 read the zero column

    // K loop: 9 a-blocks x 25 chunks of K=4. Branch-free, decode-free.
    // A frag (16x4 f32): v0 = K0 (lanes 0-15) / K2 (16-31); v1 = K1 / K3.
    v8f acc = {};
    #pragma unroll
    for (int a = 0; a < 9; ++a) {
        const float xa = xe[a];
        #pragma unroll
        for (int tc = 0; tc < NCH; ++tc) {
            const int p  = tc * 2 + hi;                       // pair index
            const int pi = p * 128 + (p & 1) * 32 + nloc * 2; // sPP dwords
            const int wi = ((a * 50 + p) * NCOL + mc) * 2;    // sW dwords
            const v2f pp = *(const v2f*)&sPP[pi];
            const v2f wv = *(const v2f*)&sW[wi];
            v2f A; A.x = xa * pp.x; A.y = xa * pp.y;
            acc = __builtin_amdgcn_wmma_f32_16x16x4_f32(
                /*neg_a=*/false, A, /*neg_b=*/false, wv,
                /*c_mod=*/(short)0, acc, /*reuse_a=*/false, /*reuse_b=*/false);
        }
    }

    // D layout: VGPR r -> row r (lanes 0-15) / r+8 (lanes 16-31), col = m.
    if (m < DB) {
        #pragma unroll
        for (int r = 0; r < 8; ++r)
            out[((size_t)(n0 + r + hi * 8) * CB + c) * DB + m] = acc[r];
    }
}

// ---------------------------------------------------------------------------
extern "C" void kernel_launch(void* const* d_in, const int* in_sizes, int n_in,
                              void* d_out, int out_size, void* d_ws, size_t ws_size,
                              hipStream_t stream)
{
    // setup_inputs() dict order: x, then (u, w) pairs per (order, irrep).
    const float* xin   = (const float*)d_in[0];
    const float* u1_0e = (const float*)d_in[1];  const float* w1_0e = (const float*)d_in[2];
    const float* u1_1o = (const float*)d_in[3];  const float* w1_1o = (const float*)d_in[4];
    const float* u1_2e = (const float*)d_in[5];  const float* w1_2e = (const float*)d_in[6];
    const float* u2_0e = (const float*)d_in[7];  const float* w2_0e = (const float*)d_in[8];
    const float* u2_1o = (const float*)d_in[9];  const float* w2_1o = (const float*)d_in[10];
    const float* u2_2e = (const float*)d_in[11]; const float* w2_2e = (const float*)d_in[12];
    const float* u3_0e = (const float*)d_in[13]; const float* w3_0e = (const float*)d_in[14];
    const float* u3_1o = (const float*)d_in[15]; const float* w3_1o = (const float*)d_in[16];
    const float* u3_2e = (const float*)d_in[17]; const float* w3_2e = (const float*)d_in[18];

    float* W   = (float*)d_ws;   // needs CB*WCH*4 = 4.61 MB of scratch
    float* out = (float*)d_out;

    const int total = CB * NFP * NCOL;   // 1,152,000
    build_weights<<<(total + 255) / 256, 256, 0, stream>>>(
        u1_0e, w1_0e, u1_1o, w1_1o, u1_2e, w1_2e,
        u2_0e, w2_0e, u2_1o, w2_1o, u2_2e, w2_2e,
        u3_0e, w3_0e, u3_1o, w3_1o, u3_2e, w3_2e, W);

    dim3 grid(NB / NROW, CB);
    stp_main<<<grid, BLK, 0, stream>>>(xin, W, out);
}